// TokenChoiceTopKRouter_82678120448635
// MI455X (gfx1250) — compile-verified
//
#include <hip/hip_runtime.h>
#include <hip/hip_bf16.h>
#include <math.h>

typedef float v2f __attribute__((ext_vector_type(2)));
typedef float v8f __attribute__((ext_vector_type(8)));

#define D_DIM 4096
#define E_DIM 64
#define TOPK  8
#define KC    64     // K-chunk staged through LDS
#define PAD   68     // LDS row stride in floats: 272B -> 16B aligned, conflict-free

// ---------------------------------------------------------------------------
// Kernel 1: logits[n, e] = sum_d x[n, d] * gate_w[e, d]
// Block: 256 threads = 8 waves. Block tile: 32 tokens x 64 experts.
// Wave w: token sub-tile (w&1)*16, expert tile (w>>1)*16.
// WMMA: V_WMMA_F32_16X16X4_F32, A = x-tile (16x4), B = gate-tile^T (4x16).
// ---------------------------------------------------------------------------
__global__ __launch_bounds__(256) void moe_gate_gemm(
    const float* __restrict__ x,
    const float* __restrict__ gw,
    float* __restrict__ logits) {
  __shared__ float xs[32 * PAD];   // 32 tokens  x KC
  __shared__ float ws[64 * PAD];   // 64 experts x KC

  const int tokenBase  = blockIdx.x * 32;
  const int tid        = threadIdx.x;
  const int wave       = tid >> 5;
  const int lane       = tid & 31;
  const int tokenSub   = (wave & 1) * 16;   // 0 or 16
  const int expertBase = (wave >> 1) * 16;  // 0,16,32,48
  const int lrow       = lane & 15;         // matrix row (A) / col (B,C)
  const int kpair      = (lane >> 4) * 2;   // K pair selector per ISA layout

  v8f acc = {};

  const int ldr = tid >> 4;         // 0..15 : staging row
  const int ldc = (tid & 15) * 4;   // 0..60 : staging col (float4)

  for (int kc = 0; kc < D_DIM; kc += KC) {
    // --- stage x tile: 32 rows x 64 cols (2 passes of 16 rows) ---
#pragma unroll
    for (int it = 0; it < 2; ++it) {
      const int row = ldr + it * 16;
      const float4 v =
          *(const float4*)(x + (size_t)(tokenBase + row) * D_DIM + kc + ldc);
      *(float4*)(xs + row * PAD + ldc) = v;
    }
    // --- stage gate_w tile: 64 rows x 64 cols (4 passes of 16 rows) ---
#pragma unroll
    for (int it = 0; it < 4; ++it) {
      const int row = ldr + it * 16;
      const float4 v = *(const float4*)(gw + (size_t)row * D_DIM + kc + ldc);
      *(float4*)(ws + row * PAD + ldc) = v;
    }
    __syncthreads();

    // --- 16 WMMA steps over the K-chunk ---
#pragma unroll
    for (int k = 0; k < KC; k += 4) {
      // A 16x4 f32: lanes 0-15 rows M=0..15 K={0,1}; lanes 16-31 K={2,3}
      v2f a = *(const v2f*)(xs + (tokenSub + lrow) * PAD + k + kpair);
      // B 4x16 f32: lanes 0-15 cols N=0..15 K={0,1}; lanes 16-31 K={2,3}
      v2f b = *(const v2f*)(ws + (expertBase + lrow) * PAD + k + kpair);
      acc = __builtin_amdgcn_wmma_f32_16x16x4_f32(
          /*neg_a=*/false, a, /*neg_b=*/false, b,
          /*c_mod=*/(short)0, acc, /*reuse_a=*/false, /*reuse_b=*/false);
    }
    __syncthreads();
  }

  // C/D 16x16 f32 layout: lane<16 -> N=lane, M=v; lane>=16 -> N=lane-16, M=8+v
  const int col    = expertBase + lrow;
  const int rowOff = (lane >> 4) * 8;
#pragma unroll
  for (int v = 0; v < 8; ++v) {
    logits[(size_t)(tokenBase + tokenSub + rowOff + v) * E_DIM + col] = acc[v];
  }
}

// ---------------------------------------------------------------------------
// Kernel 2: per-token sigmoid + biased top-8 (gather ORIGINAL scores),
// normalize, write scores + indices (as float), histogram via atomics.
// ---------------------------------------------------------------------------
__global__ __launch_bounds__(256) void router_topk(
    const float* __restrict__ logits,
    const float* __restrict__ ebias,
    float* __restrict__ out_scores,
    float* __restrict__ out_idx,
    int* __restrict__ counts, int N) {
  const int n = blockIdx.x * blockDim.x + threadIdx.x;
  if (n >= N) return;

  float s[E_DIM];
#pragma unroll 8
  for (int e = 0; e < E_DIM; ++e) {
    const float l = logits[(size_t)n * E_DIM + e];
    s[e] = 1.0f / (1.0f + __expf(-l));
  }

  unsigned long long taken = 0ull;
  float sel[TOPK];
  int   selIdx[TOPK];
  float sum = 0.0f;

#pragma unroll
  for (int k = 0; k < TOPK; ++k) {
    float best = -INFINITY;
    int   bi   = 0;
    for (int e = 0; e < E_DIM; ++e) {
      const float v  = s[e] + ebias[e];           // biased score for selection
      const bool  ok = ((taken >> e) & 1ull) == 0ull;
      if (ok && v > best) { best = v; bi = e; }
    }
    taken |= (1ull << bi);
    sel[k]    = s[bi];                            // gather ORIGINAL score
    selIdx[k] = bi;
    sum += s[bi];
    atomicAdd(&counts[bi], 1);
  }

  const float inv = 1.0f / (sum + 1e-20f);        // ROUTE_NORM
#pragma unroll
  for (int k = 0; k < TOPK; ++k) {
    out_scores[(size_t)n * TOPK + k] = sel[k] * inv;  // ROUTE_SCALE == 1.0
    out_idx[(size_t)n * TOPK + k]    = (float)selIdx[k];
  }
}

// ---------------------------------------------------------------------------
// Kernel 0/3: zero histogram (graph-replay safe) and clamp-to->=8 finalize.
// ---------------------------------------------------------------------------
__global__ void zero_counts(int* __restrict__ counts) {
  if (threadIdx.x < E_DIM) counts[threadIdx.x] = 0;
}

__global__ void finalize_counts(const int* __restrict__ counts,
                                float* __restrict__ out_counts) {
  if (threadIdx.x < E_DIM) {
    const float c = (float)counts[threadIdx.x];
    out_counts[threadIdx.x] = fmaxf(c, 8.0f);
  }
}

// ---------------------------------------------------------------------------
extern "C" void kernel_launch(void* const* d_in, const int* in_sizes, int n_in,
                              void* d_out, int out_size, void* d_ws,
                              size_t ws_size, hipStream_t stream) {
  const float* x  = (const float*)d_in[0];   // [N, 4096] fp32
  const float* gw = (const float*)d_in[1];   // [64, 4096] fp32
  const float* eb = (const float*)d_in[2];   // [64] fp32

  const int N = in_sizes[0] / D_DIM;         // 32768

  // Workspace: logits [N,64] f32, then 64 int counters.
  float* logits = (float*)d_ws;
  int*   counts = (int*)((char*)d_ws + (size_t)N * E_DIM * sizeof(float));

  // Output (flat fp32): scores [N,8] | indices [N,8] | counts [64]
  float* out_scores = (float*)d_out;
  float* out_idx    = out_scores + (size_t)N * TOPK;
  float* out_counts = out_idx + (size_t)N * TOPK;

  zero_counts<<<1, 64, 0, stream>>>(counts);
  moe_gate_gemm<<<N / 32, 256, 0, stream>>>(x, gw, logits);
  router_topk<<<(N + 255) / 256, 256, 0, stream>>>(logits, eb, out_scores,
                                                   out_idx, counts, N);
  finalize_counts<<<1, 64, 0, stream>>>(counts, out_counts);
}